// GINLayer_27453430956621
// MI455X (gfx1250) — compile-verified
//
#include <hip/hip_runtime.h>
#include <hip/hip_bf16.h>

typedef __attribute__((ext_vector_type(2))) float v2f;
typedef __attribute__((ext_vector_type(8))) float v8f;

#define IN_DIM 128
#define LDP 132   // LDS row pad: 132 % 64 banks -> conflict-free column walks

// ---------------------------------------------------------------- zero scratch
__global__ __launch_bounds__(256) void gin_zero_ws(float* __restrict__ p, size_t n) {
    size_t i = (size_t)blockIdx.x * blockDim.x + threadIdx.x;
    size_t stride = (size_t)gridDim.x * blockDim.x;
    for (; i < n; i += stride) p[i] = 0.0f;
}

// ------------------------------------------------- edge pass 1: scatter raw data
// aggA[dst] += edge_attr[e];  xs[dst] += x[src];  deg[dst] += 1
// one wave per edge, float4 per lane (4 channels), atomics land in L2-resident 5MB buffers
__global__ __launch_bounds__(256) void gin_edge_pass1(
    const int* __restrict__ ei, const float* __restrict__ edge_attr,
    const float* __restrict__ x,
    float* __restrict__ aggA, float* __restrict__ xs, float* __restrict__ deg,
    int E)
{
    int gw = (int)(((size_t)blockIdx.x * blockDim.x + threadIdx.x) >> 5);
    if (gw >= E) return;
    int lane = threadIdx.x & 31;
    int s = ei[gw];
    int d = ei[E + gw];
    const float4 ea = *(const float4*)(edge_attr + (size_t)gw * IN_DIM + lane * 4);
    const float4 xv = *(const float4*)(x + (size_t)s * IN_DIM + lane * 4);
    float* pa = aggA + (size_t)d * IN_DIM + lane * 4;
    float* px = xs   + (size_t)d * IN_DIM + lane * 4;
    atomicAdd(pa + 0, ea.x); atomicAdd(pa + 1, ea.y);
    atomicAdd(pa + 2, ea.z); atomicAdd(pa + 3, ea.w);
    atomicAdd(px + 0, xv.x); atomicAdd(px + 1, xv.y);
    atomicAdd(px + 2, xv.z); atomicAdd(px + 3, xv.w);
    if (lane == 0) atomicAdd(deg + d, 1.0f);
}

// ------------------------------------------------- edge pass 2: second-hop scatter
// S2[dst] += aggA[src];  dsum[dst] += deg[src]   (needs pass-1 complete)
__global__ __launch_bounds__(256) void gin_edge_pass2(
    const int* __restrict__ ei,
    const float* __restrict__ aggA, const float* __restrict__ deg,
    float* __restrict__ S2, float* __restrict__ dsum, int E)
{
    int gw = (int)(((size_t)blockIdx.x * blockDim.x + threadIdx.x) >> 5);
    if (gw >= E) return;
    int lane = threadIdx.x & 31;
    int s = ei[gw];
    int d = ei[E + gw];
    const float4 av = *(const float4*)(aggA + (size_t)s * IN_DIM + lane * 4);
    float* ps = S2 + (size_t)d * IN_DIM + lane * 4;
    atomicAdd(ps + 0, av.x); atomicAdd(ps + 1, av.y);
    atomicAdd(ps + 2, av.z); atomicAdd(ps + 3, av.w);
    if (lane == 0) atomicAdd(dsum + d, deg[s]);
}

// ------------------------------------------------- fused node kernel (all GEMMs via f32 WMMA)
// block = 256 threads = 8 waves; tile = 16 nodes; each wave owns a 16-col slab.
// h2 = (aggA+S2)@We + (deg+dsum)*be ; r = relu(h1@W1t + h2@W1b + b1) ; out = r@W2 + b2
__global__ __launch_bounds__(256) void gin_node_kernel(
    const float* __restrict__ x,
    const float* __restrict__ We, const float* __restrict__ be,
    const float* __restrict__ W1, const float* __restrict__ b1,
    const float* __restrict__ W2, const float* __restrict__ b2,
    const float* __restrict__ aggA, const float* __restrict__ S2,
    const float* __restrict__ xs,
    const float* __restrict__ deg, const float* __restrict__ dsum,
    float* __restrict__ out, int N)
{
    __shared__ float hA[16][LDP];
    __shared__ float h1[16][LDP];
    __shared__ float h2[16][LDP];
    __shared__ float rb[16][LDP];
    __shared__ float ds16[16];

    const int tid = threadIdx.x;
    const int tile0 = blockIdx.x * 16;

    // cooperative load of per-node activations into LDS
    for (int idx = tid; idx < 16 * IN_DIM; idx += 256) {
        int row = idx >> 7, colc = idx & 127;
        int node = tile0 + row;
        float a = 0.0f, b = 0.0f;
        if (node < N) {
            size_t off = (size_t)node * IN_DIM + colc;
            a = aggA[off] + S2[off];
            b = x[off] + xs[off];
        }
        hA[row][colc] = a;
        h1[row][colc] = b;
    }
    if (tid < 16) {
        int node = tile0 + tid;
        ds16[tid] = (node < N) ? (deg[node] + dsum[node]) : 0.0f;
    }
    __syncthreads();

    const int lane = tid & 31;
    const int wid  = tid >> 5;
    const int nr   = lane & 15;       // A-row / B,D-column within tile
    const int half = lane >> 4;       // K/M sub-group select
    const int col  = wid * 16 + nr;   // output column 0..127

    // ---- stage 1: h2 = hA @ We + ds*be
    v8f c = {};
    for (int k0 = 0; k0 < IN_DIM; k0 += 4) {
        int ka = k0 + 2 * half;
        v2f a = *(const v2f*)&hA[nr][ka];
        v2f b;
        b.x = We[(size_t)(ka    ) * IN_DIM + col];
        b.y = We[(size_t)(ka + 1) * IN_DIM + col];
        c = __builtin_amdgcn_wmma_f32_16x16x4_f32(false, a, false, b, (short)0, c, false, false);
    }
    {
        float ben = be[col];
        #pragma unroll
        for (int r = 0; r < 8; ++r) {
            int m = r + 8 * half;
            h2[m][col] = c[r] + ds16[m] * ben;
        }
    }
    __syncthreads();

    // ---- stage 2: rb = relu(h1@W1[0:128] + h2@W1[128:256] + b1)
    v8f c2 = {};
    for (int k0 = 0; k0 < IN_DIM; k0 += 4) {
        int ka = k0 + 2 * half;
        v2f a = *(const v2f*)&h1[nr][ka];
        v2f b;
        b.x = W1[(size_t)(ka    ) * IN_DIM + col];
        b.y = W1[(size_t)(ka + 1) * IN_DIM + col];
        c2 = __builtin_amdgcn_wmma_f32_16x16x4_f32(false, a, false, b, (short)0, c2, false, false);
    }
    for (int k0 = 0; k0 < IN_DIM; k0 += 4) {
        int ka = k0 + 2 * half;
        v2f a = *(const v2f*)&h2[nr][ka];
        v2f b;
        b.x = W1[(size_t)(IN_DIM + ka    ) * IN_DIM + col];
        b.y = W1[(size_t)(IN_DIM + ka + 1) * IN_DIM + col];
        c2 = __builtin_amdgcn_wmma_f32_16x16x4_f32(false, a, false, b, (short)0, c2, false, false);
    }
    {
        float b1n = b1[col];
        #pragma unroll
        for (int r = 0; r < 8; ++r) {
            int m = r + 8 * half;
            rb[m][col] = fmaxf(c2[r] + b1n, 0.0f);
        }
    }
    __syncthreads();

    // ---- stage 3: out = rb @ W2 + b2
    v8f c3 = {};
    for (int k0 = 0; k0 < IN_DIM; k0 += 4) {
        int ka = k0 + 2 * half;
        v2f a = *(const v2f*)&rb[nr][ka];
        v2f b;
        b.x = W2[(size_t)(ka    ) * IN_DIM + col];
        b.y = W2[(size_t)(ka + 1) * IN_DIM + col];
        c3 = __builtin_amdgcn_wmma_f32_16x16x4_f32(false, a, false, b, (short)0, c3, false, false);
    }
    {
        float b2n = b2[col];
        #pragma unroll
        for (int r = 0; r < 8; ++r) {
            int m = r + 8 * half;
            int node = tile0 + m;
            if (node < N) out[(size_t)node * IN_DIM + col] = c3[r] + b2n;
        }
    }
}

extern "C" void kernel_launch(void* const* d_in, const int* in_sizes, int n_in,
                              void* d_out, int out_size, void* d_ws, size_t ws_size,
                              hipStream_t stream) {
    const float* x         = (const float*)d_in[0];
    const int*   ei        = (const int*)d_in[1];   // [2][E]: src row, dst row
    const float* edge_attr = (const float*)d_in[2];
    const float* We        = (const float*)d_in[3];
    const float* be        = (const float*)d_in[4];
    const float* W1        = (const float*)d_in[5];
    const float* b1        = (const float*)d_in[6];
    const float* W2        = (const float*)d_in[7];
    const float* b2        = (const float*)d_in[8];
    float*       out       = (float*)d_out;

    const int N = in_sizes[0] / IN_DIM;
    const int E = in_sizes[1] / 2;

    // workspace layout (floats): aggA[N*128] | xs[N*128] | S2[N*128] | deg[N] | dsum[N]
    float* ws   = (float*)d_ws;
    float* aggA = ws;
    float* xs   = aggA + (size_t)N * IN_DIM;
    float* S2   = xs   + (size_t)N * IN_DIM;
    float* deg  = S2   + (size_t)N * IN_DIM;
    float* dsum = deg  + N;
    size_t zcount = (size_t)N * (3 * IN_DIM) + 2 * (size_t)N;

    gin_zero_ws<<<2048, 256, 0, stream>>>(ws, zcount);

    int ep_blocks = (E + 7) / 8;  // 8 waves (edges) per 256-thread block
    gin_edge_pass1<<<ep_blocks, 256, 0, stream>>>(ei, edge_attr, x, aggA, xs, deg, E);
    gin_edge_pass2<<<ep_blocks, 256, 0, stream>>>(ei, aggA, deg, S2, dsum, E);

    int nb = (N + 15) / 16;
    gin_node_kernel<<<nb, 256, 0, stream>>>(x, We, be, W1, b1, W2, b2,
                                            aggA, S2, xs, deg, dsum, out, N);
}